// LocalConv1D_4355096838438
// MI455X (gfx1250) — compile-verified
//
#include <hip/hip_runtime.h>

typedef float v2f __attribute__((ext_vector_type(2)));
typedef float v8f __attribute__((ext_vector_type(8)));

#define IN_CH   32
#define OUT_CH  32
#define BATCH   16
#define NT      32           // nodes per workgroup
#define THREADS 256          // 8 wave32
#define WAVES   8
#define NPW     (NT / WAVES) // nodes per wave = 4

// LDS layout ------------------------------------------------------------
// x tile stored as float2 pairs (h even/odd) so each B-matrix fetch is one
// ds_load_b64: row = (h/2)*BATCH + b  (256 rows), pitch 33 float2 (66 floats)
// -> byte stride 264/row: lanes 0-15 hit banks {2*lid}, lanes 16-31 (h2+1)
//    hit banks {32+2*lid}: zero conflicts, all 64 banks used.
#define XPITCH   66                       // floats per x row
#define XS_FLOATS ((IN_CH/2)*BATCH*XPITCH)   // 256*66 = 16896 floats
// out tile: idx = b*OB + o*NT + j ; OB = 32*32+1 so the 8 per-lane D stores
// (lanes 0-15 distinct banks, lane/lane+16 2-way) and the coalesced
// write-out (consecutive j) are conflict-(mostly)free.
#define OB       (OUT_CH*NT + 1)             // 1025
#define OS_FLOATS ((BATCH-1)*OB + (OUT_CH-1)*NT + NT)  // 16399 floats
#define SMEM_BYTES ((XS_FLOATS + OS_FLOATS) * sizeof(float))  // ~133 KB

__global__ __launch_bounds__(THREADS)
void LocalConv1D_wmma_kernel(const float* __restrict__ x,
                             const float* __restrict__ W,
                             const float* __restrict__ bias,
                             float* __restrict__ out,
                             int nNodes)
{
    extern __shared__ float smem[];
    float* xs = smem;               // staged x tile
    float* os = smem + XS_FLOATS;   // staged out tile

    const int t  = threadIdx.x;
    const int n0 = blockIdx.x * NT;

    // ---- Stage x[b,h,n0..n0+31] -> LDS, fully coalesced 128B per wave ----
    for (int i = t; i < BATCH * IN_CH * NT; i += THREADS) {
        const int j  = i & (NT - 1);
        const int bh = i >> 5;              // NT == 32
        const int h  = bh & (IN_CH - 1);
        const int b  = bh >> 5;             // IN_CH == 32
        const int n  = n0 + j;
        float v = 0.0f;
        if (n < nNodes)
            v = x[((size_t)b * IN_CH + h) * (size_t)nNodes + n];
        xs[((h >> 1) * BATCH + b) * XPITCH + j * 2 + (h & 1)] = v;
    }
    __syncthreads();

    // ---- Per-node 32x32 * 32x16 GEMMs via V_WMMA_F32_16X16X4_F32 ----
    const int lane = t & 31;
    const int wave = t >> 5;
    const int half = lane >> 4;   // 0: lanes 0-15, 1: lanes 16-31
    const int lid  = lane & 15;

    for (int i = 0; i < NPW; ++i) {
        const int jn = wave * NPW + i;      // local node (wave-uniform)
        const int nd = n0 + jn;
        if (nd < nNodes) {                  // wave-uniform branch: EXEC stays full
            const float* Wn = W    + (size_t)nd * (OUT_CH * IN_CH);
            const float* bn = bias + (size_t)nd * OUT_CH;

            // init accumulators with bias broadcast along batch (N) dim:
            // C/D layout: lane holds (M = r + 8*half, N = lid), M == out ch
            v8f acc0, acc1;
            #pragma unroll
            for (int r = 0; r < 8; ++r) {
                acc0[r] = bn[ 0 + half * 8 + r];
                acc1[r] = bn[16 + half * 8 + r];
            }

            if (i + 1 < NPW)   // prefetch next node's 4KB weight block
                __builtin_prefetch(Wn + OUT_CH * IN_CH, 0, 1);

            #pragma unroll
            for (int k0 = 0; k0 < IN_CH; k0 += 4) {
                const int kk = k0 + 2 * half;          // even -> 8B aligned

                // B (4x16, K x batch): one ds_load_b64 per lane
                v2f bv = *(const v2f*)&xs[((kk >> 1) * BATCH + lid) * XPITCH + jn * 2];

                // A (16x4) per o-tile: lane m holds W[o=m, kk], W[o=m, kk+1]
                v2f a0 = *(const v2f*)(Wn + ( 0 + lid) * IN_CH + kk);
                v2f a1 = *(const v2f*)(Wn + (16 + lid) * IN_CH + kk);

                acc0 = __builtin_amdgcn_wmma_f32_16x16x4_f32(
                           false, a0, false, bv, (short)0, acc0, false, false);
                acc1 = __builtin_amdgcn_wmma_f32_16x16x4_f32(
                           false, a1, false, bv, (short)0, acc1, false, false);
            }

            // D -> LDS out tile: lane holds (o = r + 8*half [+16], b = lid)
            #pragma unroll
            for (int r = 0; r < 8; ++r) {
                const int o = r + 8 * half;
                os[lid * OB + (o     ) * NT + jn] = acc0[r];
                os[lid * OB + (o + 16) * NT + jn] = acc1[r];
            }
        }
    }
    __syncthreads();

    // ---- Coalesced write-out: out[b,o,n0..n0+31], 128B per wave ----
    for (int i = t; i < BATCH * OUT_CH * NT; i += THREADS) {
        const int j  = i & (NT - 1);
        const int bo = i >> 5;
        const int o  = bo & (OUT_CH - 1);
        const int b  = bo >> 5;
        const int n  = n0 + j;
        if (n < nNodes)
            out[((size_t)b * OUT_CH + o) * (size_t)nNodes + n] =
                os[b * OB + o * NT + j];
    }
}

extern "C" void kernel_launch(void* const* d_in, const int* in_sizes, int n_in,
                              void* d_out, int out_size, void* d_ws, size_t ws_size,
                              hipStream_t stream) {
    const float* x    = (const float*)d_in[0];
    const float* W    = (const float*)d_in[1];
    const float* bias = (const float*)d_in[2];
    float*       out  = (float*)d_out;

    const int nNodes = in_sizes[2] / OUT_CH;   // bias is [N, OUT_CH]
    const int grid   = (nNodes + NT - 1) / NT;

    LocalConv1D_wmma_kernel<<<grid, THREADS, SMEM_BYTES, stream>>>(
        x, W, bias, out, nNodes);
}